// APPNP_NC_43542378447169
// MI455X (gfx1250) — compile-verified
//
#include <hip/hip_runtime.h>
#include <hip/hip_bf16.h>

#define N_NODES   100000
#define N_EDGES   3200000
#define N_FEATS   512
#define HIDDEN    256
#define N_CLASSES 64
#define K_STEPS   10
#define ALPHA     0.1f

typedef __attribute__((ext_vector_type(2))) float v2f;
typedef __attribute__((ext_vector_type(8))) float v8f;

// Padded LDS strides (floats). 516 % 64 banks -> stride-4 bank pattern, conflict free.
#define XS_LD (N_FEATS + 4)   // 516, multiple of 4 so float4 stores stay aligned
#define HS_LD (HIDDEN + 4)    // 260

// ---------------------------------------------------------------------------
// Fused MLP: h0 = relu(x @ W1 + b1) @ W2 + b2, written to h0 and hcur.
// One block = 16 node rows. 256 threads = 8 waves. f32 WMMA 16x16x4.
// ---------------------------------------------------------------------------
__launch_bounds__(256)
__global__ void mlp_fused_kernel(const float* __restrict__ x,
                                 const float* __restrict__ W1,
                                 const float* __restrict__ b1,
                                 const float* __restrict__ W2,
                                 const float* __restrict__ b2,
                                 float* __restrict__ h0,
                                 float* __restrict__ hcur) {
    __shared__ float xs[16 * XS_LD];  // 16x512 x-tile (padded)
    __shared__ float hs[16 * HS_LD];  // 16x256 hidden tile (padded)

    const int tid  = threadIdx.x;
    const int wave = tid >> 5;
    const int lane = tid & 31;
    const int hf   = lane >> 4;   // which 16-lane half
    const int l16  = lane & 15;
    const long m0  = (long)blockIdx.x * 16;

    // Cooperative load of the 16x512 input tile (float4, 2048 vectors).
    {
        const float4* xin = (const float4*)(x + m0 * N_FEATS);
        for (int i = tid; i < 16 * (N_FEATS / 4); i += 256) {
            const int row  = i >> 7;          // 128 float4 per row
            const int col4 = i & 127;
            *(float4*)(&xs[row * XS_LD + col4 * 4]) = xin[row * (N_FEATS / 4) + col4];
        }
    }
    __syncthreads();

    // GEMM1: 16x512 @ 512x256. Each wave owns n-tiles {wave, wave+8}.
    for (int t = 0; t < 2; ++t) {
        const int n0 = (wave + t * 8) * 16;
        v8f c = {};
        for (int k = 0; k < N_FEATS; k += 4) {
            // A fragment (16x4 f32): lanes 0-15 K={k,k+1}, lanes 16-31 K={k+2,k+3}
            v2f a, b;
            a.x = xs[l16 * XS_LD + k + 2 * hf + 0];
            a.y = xs[l16 * XS_LD + k + 2 * hf + 1];
            // B fragment (4x16 f32): row-striped across lanes
            b.x = W1[(k + 2 * hf + 0) * HIDDEN + n0 + l16];
            b.y = W1[(k + 2 * hf + 1) * HIDDEN + n0 + l16];
            c = __builtin_amdgcn_wmma_f32_16x16x4_f32(
                    false, a, false, b, (short)0, c, false, false);
        }
        // bias + ReLU -> LDS hidden tile. VGPR v holds rows {v, v+8} per half.
        const float bias = b1[n0 + l16];
        for (int v = 0; v < 8; ++v) {
            float r = c[v] + bias;
            hs[(v + 8 * hf) * HS_LD + n0 + l16] = r > 0.0f ? r : 0.0f;
        }
    }
    __syncthreads();

    // GEMM2: 16x256 @ 256x64. Waves 0..3 each own one 16-col tile.
    if (wave < 4) {
        const int n0 = wave * 16;
        v8f c = {};
        for (int k = 0; k < HIDDEN; k += 4) {
            v2f a, b;
            a.x = hs[l16 * HS_LD + k + 2 * hf + 0];
            a.y = hs[l16 * HS_LD + k + 2 * hf + 1];
            b.x = W2[(k + 2 * hf + 0) * N_CLASSES + n0 + l16];
            b.y = W2[(k + 2 * hf + 1) * N_CLASSES + n0 + l16];
            c = __builtin_amdgcn_wmma_f32_16x16x4_f32(
                    false, a, false, b, (short)0, c, false, false);
        }
        const float bias = b2[n0 + l16];
        for (int v = 0; v < 8; ++v) {
            const float r   = c[v] + bias;
            const long  row = m0 + v + 8 * hf;
            h0[row * N_CLASSES + n0 + l16]   = r;
            hcur[row * N_CLASSES + n0 + l16] = r;
        }
    }
}

// ---------------------------------------------------------------------------
// Degree count over dst edges (self-loop added in dinv kernel).
// ---------------------------------------------------------------------------
__launch_bounds__(256)
__global__ void deg_kernel(const long long* __restrict__ dst,
                           float* __restrict__ deg) {
    const int e = blockIdx.x * 256 + threadIdx.x;
    if (e < N_EDGES) atomicAdd(&deg[(int)dst[e]], 1.0f);
}

__launch_bounds__(256)
__global__ void dinv_kernel(const float* __restrict__ deg,
                            float* __restrict__ dinv) {
    const int i = blockIdx.x * 256 + threadIdx.x;
    if (i < N_NODES) {
        const float d = deg[i] + 1.0f;   // +1 self loop; always > 0
        dinv[i] = rsqrtf(d);
    }
}

__launch_bounds__(256)
__global__ void norm_kernel(const long long* __restrict__ src,
                            const long long* __restrict__ dst,
                            const float* __restrict__ dinv,
                            float* __restrict__ nrm) {
    const int e = blockIdx.x * 256 + threadIdx.x;
    if (e < N_EDGES) nrm[e] = dinv[(int)src[e]] * dinv[(int)dst[e]];
}

// ---------------------------------------------------------------------------
// One wave per edge: agg[dst] += norm * hcur[src]  (64 f32 = float2 per lane)
// ---------------------------------------------------------------------------
__launch_bounds__(256)
__global__ void scatter_kernel(const long long* __restrict__ src,
                               const long long* __restrict__ dst,
                               const float* __restrict__ nrm,
                               const float* __restrict__ hcur,
                               float* __restrict__ agg) {
    const int e = blockIdx.x * 8 + (threadIdx.x >> 5);
    if (e >= N_EDGES) return;
    const int  lane = threadIdx.x & 31;
    const long s = (long)src[e];
    const long d = (long)dst[e];
    const float w = nrm[e];
    const float2 hv = ((const float2*)(hcur + s * N_CLASSES))[lane];
    float* ap = agg + d * N_CLASSES + lane * 2;
    atomicAdd(ap + 0, w * hv.x);
    atomicAdd(ap + 1, w * hv.y);
}

// ---------------------------------------------------------------------------
// hcur = (1-a)*(agg + dinv^2 * hcur) + a*h0   (self loop folded in)
// ---------------------------------------------------------------------------
__launch_bounds__(256)
__global__ void combine_kernel(const float* __restrict__ agg,
                               const float* __restrict__ h0,
                               const float* __restrict__ dinv,
                               float* __restrict__ hcur) {
    const long idx = (long)blockIdx.x * 256 + threadIdx.x;
    if (idx >= (long)N_NODES * N_CLASSES) return;
    const int  node = (int)(idx >> 6);   // /N_CLASSES
    const float di  = dinv[node];
    const float self = di * di * hcur[idx];
    hcur[idx] = (1.0f - ALPHA) * (agg[idx] + self) + ALPHA * h0[idx];
}

// ---------------------------------------------------------------------------
// log_softmax per node row (one wave per node, float2 per lane, wave32 shfl)
// ---------------------------------------------------------------------------
__launch_bounds__(256)
__global__ void logsoftmax_kernel(const float* __restrict__ hcur,
                                  float* __restrict__ out) {
    const int node = blockIdx.x * 8 + (threadIdx.x >> 5);
    if (node >= N_NODES) return;
    const int lane = threadIdx.x & 31;
    const float2 v = ((const float2*)(hcur + (long)node * N_CLASSES))[lane];

    float m = fmaxf(v.x, v.y);
    #pragma unroll
    for (int o = 16; o > 0; o >>= 1) m = fmaxf(m, __shfl_xor(m, o, 32));

    float s = expf(v.x - m) + expf(v.y - m);
    #pragma unroll
    for (int o = 16; o > 0; o >>= 1) s += __shfl_xor(s, o, 32);

    const float lse = m + logf(s);
    float2 r;
    r.x = v.x - lse;
    r.y = v.y - lse;
    ((float2*)(out + (long)node * N_CLASSES))[lane] = r;
}

// ---------------------------------------------------------------------------
extern "C" void kernel_launch(void* const* d_in, const int* in_sizes, int n_in,
                              void* d_out, int out_size, void* d_ws, size_t ws_size,
                              hipStream_t stream) {
    const float*     x   = (const float*)d_in[0];
    const long long* ei  = (const long long*)d_in[1];   // int64 per reference
    const float*     W1  = (const float*)d_in[2];
    const float*     b1  = (const float*)d_in[3];
    const float*     W2  = (const float*)d_in[4];
    const float*     b2  = (const float*)d_in[5];
    float*           out = (float*)d_out;

    const long long* src = ei;            // edge_index[0]
    const long long* dst = ei + N_EDGES;  // edge_index[1]

    // Workspace carve-up (all 256B aligned).
    const size_t HN  = (size_t)N_NODES * N_CLASSES * sizeof(float);  // 25.6 MB
    const size_t NN  = (size_t)N_NODES * sizeof(float);              // 0.4 MB
    char* ws = (char*)d_ws;
    float* h0   = (float*)(ws);
    float* hcur = (float*)(ws + HN);
    float* agg  = (float*)(ws + 2 * HN);
    float* deg  = (float*)(ws + 3 * HN);
    float* dinv = (float*)(ws + 3 * HN + ((NN + 255) & ~(size_t)255));
    float* nrm  = (float*)(ws + 3 * HN + 2 * ((NN + 255) & ~(size_t)255));

    // 1) Fused MLP -> h0 (and hcur = h0).
    mlp_fused_kernel<<<N_NODES / 16, 256, 0, stream>>>(x, W1, b1, W2, b2, h0, hcur);

    // 2) Degrees, dinv, per-edge norm.
    hipMemsetAsync(deg, 0, NN, stream);
    deg_kernel<<<N_EDGES / 256, 256, 0, stream>>>(dst, deg);
    dinv_kernel<<<(N_NODES + 255) / 256, 256, 0, stream>>>(deg, dinv);
    norm_kernel<<<N_EDGES / 256, 256, 0, stream>>>(src, dst, dinv, nrm);

    // 3) APPNP propagation, K steps.
    for (int step = 0; step < K_STEPS; ++step) {
        hipMemsetAsync(agg, 0, HN, stream);
        scatter_kernel<<<N_EDGES / 8, 256, 0, stream>>>(src, dst, nrm, hcur, agg);
        combine_kernel<<<(N_NODES * N_CLASSES) / 256, 256, 0, stream>>>(agg, h0, dinv, hcur);
    }

    // 4) log_softmax -> d_out.
    logsoftmax_kernel<<<N_NODES / 8, 256, 0, stream>>>(hcur, out);
}